// TAN_14113262535395
// MI455X (gfx1250) — compile-verified
//
#include <hip/hip_runtime.h>
#include <hip/hip_bf16.h>
#include <math.h>

#define N_SENT 131072
#define N_BAGS 16384
#define DCLS   512
#define REL    53

typedef __attribute__((ext_vector_type(2))) float v2f;
typedef __attribute__((ext_vector_type(8))) float v8f;

// ---------------------------------------------------------------------------
// Kernel 1: fused per-bag attention with online softmax. One wave32 per bag.
// Reads X exactly once (268 MB dominates total traffic -> ~11.5us @ 23.3TB/s).
// Each lane keeps 16 floats (4x float4) of Con and of the accumulator in VGPRs.
// ---------------------------------------------------------------------------
__global__ __launch_bounds__(256) void tan_bag_attn(
    const float* __restrict__ X, const int* __restrict__ scope,
    const float* __restrict__ Con, const int* __restrict__ rel,
    float* __restrict__ bag)
{
  const int lane = threadIdx.x & 31;
  const int wave = threadIdx.x >> 5;
  const int b = blockIdx.x * 8 + wave;          // grid sized exactly: no guard

  const int s0 = scope[2 * b];
  const int s1 = scope[2 * b + 1];
  const float4* con4 = (const float4*)(Con + (size_t)rel[b] * DCLS);

  float4 c4[4], a4[4];
#pragma unroll
  for (int q = 0; q < 4; ++q) {
    c4[q] = con4[lane + 32 * q];
    a4[q] = make_float4(0.f, 0.f, 0.f, 0.f);
  }

  float m = -INFINITY, z = 0.f;
  for (int j = s0; j < s1; ++j) {
    const float4* x4 = (const float4*)(X + (size_t)j * DCLS);
    if (j + 1 < s1)                               // uniform branch per wave
      __builtin_prefetch((const float*)x4 + DCLS, 0, 1);  // global_prefetch_b8

    float4 xv[4];
    float s = 0.f;
#pragma unroll
    for (int q = 0; q < 4; ++q) {                 // b128 loads, coalesced
      xv[q] = x4[lane + 32 * q];
      s = fmaf(xv[q].x, c4[q].x, s);
      s = fmaf(xv[q].y, c4[q].y, s);
      s = fmaf(xv[q].z, c4[q].z, s);
      s = fmaf(xv[q].w, c4[q].w, s);
    }
    // wave32 all-reduce of the dot product
#pragma unroll
    for (int off = 16; off > 0; off >>= 1) s += __shfl_xor(s, off, 32);

    // online softmax update
    const float mn    = fmaxf(m, s);
    const float scale = __expf(m - mn);           // first iter: exp(-inf)=0
    const float w     = __expf(s - mn);
    z = z * scale + w;
#pragma unroll
    for (int q = 0; q < 4; ++q) {
      a4[q].x = fmaf(a4[q].x, scale, w * xv[q].x);
      a4[q].y = fmaf(a4[q].y, scale, w * xv[q].y);
      a4[q].z = fmaf(a4[q].z, scale, w * xv[q].z);
      a4[q].w = fmaf(a4[q].w, scale, w * xv[q].w);
    }
    m = mn;
  }

  const float inv = 1.f / z;
  float4* o4 = (float4*)(bag + (size_t)b * DCLS);
#pragma unroll
  for (int q = 0; q < 4; ++q) {
    float4 r = a4[q];
    r.x *= inv; r.y *= inv; r.z *= inv; r.w *= inv;
    o4[lane + 32 * q] = r;
  }
}

// ---------------------------------------------------------------------------
// Kernel 2: out[16384,53] = bag @ W_cls^T + b_cls via V_WMMA_F32_16X16X4_F32.
// One wave32 per 16x16 output tile; N padded 53 -> 64 (4 tiles), stores masked.
// A (16x4 f32) lane layout: lane<16 holds A[m][k0,k0+1]; lane>=16 A[m][k0+2,k0+3].
// B (4x16 f32) symmetric with N across lanes. C/D: 8 VGPRs, std 16x16 layout.
// ---------------------------------------------------------------------------
__global__ __launch_bounds__(256) void tan_classifier(
    const float* __restrict__ bag, const float* __restrict__ W,
    const float* __restrict__ bias, float* __restrict__ out)
{
  const int lane = threadIdx.x & 31;
  const int wave = threadIdx.x >> 5;
  const int gw = blockIdx.x * 8 + wave;          // 4096 waves exactly
  const int mt = gw >> 2;                        // 1024 M tiles
  const int nt = gw & 3;                         // 4 N tiles
  const int m0 = mt * 16;
  const int n0 = nt * 16;

  const int r16  = lane & 15;
  const int koff = (lane >> 4) << 1;             // 0 (lanes 0-15) or 2 (16-31)

  const float* arow = bag + (size_t)(m0 + r16) * DCLS + koff;
  const int ng  = n0 + r16;
  const int ngc = ng < REL ? ng : (REL - 1);     // clamp pad rows (v_cndmask)
  const float* brow = W + (size_t)ngc * DCLS + koff;

  v8f cacc = {};
#if __has_builtin(__builtin_amdgcn_wmma_f32_16x16x4_f32)
#pragma unroll 4
  for (int k = 0; k < DCLS; k += 4) {
    v2f a  = *(const v2f*)(arow + k);            // global_load_b64
    v2f bb = *(const v2f*)(brow + k);
    cacc = __builtin_amdgcn_wmma_f32_16x16x4_f32(
        /*neg_a=*/false, a, /*neg_b=*/false, bb,
        /*c_mod=*/(short)0, cacc, /*reuse_a=*/false, /*reuse_b=*/false);
  }
#else
  // Scalar fallback matching the WMMA C/D lane layout (should not be taken).
  {
    const int mb = m0 + ((lane >> 4) << 3);
    for (int r = 0; r < 8; ++r) {
      float acc = 0.f;
      const float* ar = bag + (size_t)(mb + r) * DCLS;
      const float* br = W + (size_t)ngc * DCLS;
      for (int k = 0; k < DCLS; ++k) acc = fmaf(ar[k], br[k], acc);
      cacc[r] = acc;
    }
  }
#endif

  if (ng < REL) {                                // mask pad columns (post-WMMA)
    const float bv = bias[ng];
    const int mb = m0 + ((lane >> 4) << 3);      // lanes>=16 hold M = 8..15
#pragma unroll
    for (int r = 0; r < 8; ++r)
      out[(size_t)(mb + r) * REL + ng] = cacc[r] + bv;
  }
}

extern "C" void kernel_launch(void* const* d_in, const int* in_sizes, int n_in,
                              void* d_out, int out_size, void* d_ws, size_t ws_size,
                              hipStream_t stream) {
  (void)in_sizes; (void)n_in; (void)out_size; (void)ws_size;
  const float* X     = (const float*)d_in[0];   // [131072, 512] f32
  const int*   scope = (const int*)  d_in[1];   // [16384, 2] i32
  const float* Con   = (const float*)d_in[2];   // [53, 512] f32
  const int*   rel   = (const int*)  d_in[3];   // [16384] i32
  const float* W     = (const float*)d_in[4];   // [53, 512] f32
  const float* bias  = (const float*)d_in[5];   // [53] f32
  float* out = (float*)d_out;                   // [16384, 53] f32
  float* bagbuf = (float*)d_ws;                 // needs 16384*512*4 = 32 MiB

  // Kernel 1: 16384 bags, 8 waves/block -> 2048 blocks (exact, no tail).
  tan_bag_attn<<<N_BAGS / 8, 256, 0, stream>>>(X, scope, Con, rel, bagbuf);
  // Kernel 2: 1024 M-tiles x 4 N-tiles = 4096 waves -> 512 blocks (exact).
  tan_classifier<<<(N_BAGS / 16) * 4 / 8, 256, 0, stream>>>(bagbuf, W, bias, out);
}